// SAGEGraph_37039797961388
// MI455X (gfx1250) — compile-verified
//
#include <hip/hip_runtime.h>
#include <math.h>

typedef __attribute__((ext_vector_type(2))) float v2f;
typedef __attribute__((ext_vector_type(8))) float v8f;

#define FEAT 128   // feature / embed width (both layers are 128-wide)

// ---------------------------------------------------------------------------
// Kernel 1: CSR offsets from sorted edge_dst. off[v] = lower_bound(dst, v),
// for v in [0, N]. deg(v) = off[v+1]-off[v]. No atomics anywhere.
// ---------------------------------------------------------------------------
__global__ __launch_bounds__(256)
void k_off(const int* __restrict__ dst, int* __restrict__ off, int nE, int nV) {
    int v = blockIdx.x * blockDim.x + threadIdx.x;
    if (v > nV) return;
    int lo = 0, hi = nE;
    while (lo < hi) {
        int mid = (lo + hi) >> 1;
        if (dst[mid] < v) lo = mid + 1; else hi = mid;
    }
    off[v] = lo;
}

// ---------------------------------------------------------------------------
// Kernel 2: segmented mean aggregation (self + in-neighbors).
// One wave32 per output row; each lane owns 4 of the 128 floats, so each edge
// is one fully-coalesced 512B row read. USE_MAP=true restricts to batch nodes.
// ---------------------------------------------------------------------------
template <bool USE_MAP>
__global__ __launch_bounds__(256)
void k_agg(const float* __restrict__ xin, const int* __restrict__ edge_src,
           const int* __restrict__ off, const int* __restrict__ map,
           float* __restrict__ out, int nout) {
    const int wave = threadIdx.x >> 5;
    const int lane = threadIdx.x & 31;
    const int i = blockIdx.x * 8 + wave;          // output row
    if (i >= nout) return;
    const int v  = USE_MAP ? map[i] : i;          // graph node
    const int e0 = off[v], e1 = off[v + 1];

    const float4* selfRow = (const float4*)(xin + (size_t)v * FEAT);
    float4 s = selfRow[lane];
    float a0 = s.x, a1 = s.y, a2 = s.z, a3 = s.w; // self-loop term

    for (int e = e0; e < e1; ++e) {
        const int src = edge_src[e];
        const float4 m = ((const float4*)(xin + (size_t)src * FEAT))[lane];
        a0 += m.x; a1 += m.y; a2 += m.z; a3 += m.w;
    }
    const float inv = 1.0f / (float)(e1 - e0 + 1);
    float4 r; r.x = a0 * inv; r.y = a1 * inv; r.z = a2 * inv; r.w = a3 * inv;
    ((float4*)(out + (size_t)i * FEAT))[lane] = r;
}

// ---------------------------------------------------------------------------
// Kernel 3: fused GEMM + bias + ReLU using fp32 WMMA (V_WMMA_F32_16X16X4_F32),
// in-place: x <- relu(x @ W + b), x is [nrows, 128], W is [128, 128].
// W staged into LDS once per block. One wave computes a 16x128 strip:
// 8 accumulator tiles, 32 K-steps of K=4, A-fragment shared across N-tiles.
// ---------------------------------------------------------------------------
__global__ __launch_bounds__(256)
void k_gemm_bias_relu(float* __restrict__ x, const float* __restrict__ W,
                      const float* __restrict__ bias, int nrows) {
    __shared__ float sW[FEAT * FEAT];             // 64 KB of 320 KB WGP LDS
    for (int i = threadIdx.x; i < (FEAT * FEAT) / 4; i += 256)
        ((float4*)sW)[i] = ((const float4*)W)[i];
    __syncthreads();

    const int wave = threadIdx.x >> 5;
    const int lane = threadIdx.x & 31;
    const int m0 = (blockIdx.x * 8 + wave) * 16;
    if (m0 >= nrows) return;
    const int half = lane >> 4;                   // 0: lanes 0-15, 1: lanes 16-31
    const int lm   = lane & 15;

    v8f acc[8] = {};                              // 8 tiles of 16x16 fp32
    const int arow = min(m0 + lm, nrows - 1);     // A row this lane supplies
    const float* aptr = x + (size_t)arow * FEAT;

    for (int kk = 0; kk < 32; ++kk) {
        const int kb = kk * 4 + 2 * half;         // K base for this lane-half
        v2f a; a.x = aptr[kb]; a.y = aptr[kb + 1];          // A 16x4 fragment
        #pragma unroll
        for (int nt = 0; nt < 8; ++nt) {
            const int n = nt * 16 + lm;
            v2f bf;                                          // B 4x16 fragment
            bf.x = sW[(kb + 0) * FEAT + n];
            bf.y = sW[(kb + 1) * FEAT + n];
            acc[nt] = __builtin_amdgcn_wmma_f32_16x16x4_f32(
                false, a, false, bf, (short)0, acc[nt], false, false);
        }
    }

    // C/D layout: VGPR r -> row m0 + r + 8*half, col = nt*16 + lm.
    #pragma unroll
    for (int r = 0; r < 8; ++r) {
        const int row = m0 + r + 8 * half;
        if (row >= nrows) continue;
        #pragma unroll
        for (int nt = 0; nt < 8; ++nt) {
            const int n = nt * 16 + lm;
            x[(size_t)row * FEAT + n] = fmaxf(acc[nt][r] + bias[n], 0.0f);
        }
    }
}

// ---------------------------------------------------------------------------
// Kernel 4: classifier logits + log-softmax. One wave per batch row.
// Lane l owns class l, and class 32+l when l<8 (C=40). Wave32 xor-shuffle
// reductions for max and sum(exp).
// ---------------------------------------------------------------------------
__global__ __launch_bounds__(256)
void k_logits(const float* __restrict__ x, const float* __restrict__ Wl,
              const float* __restrict__ bl, float* __restrict__ out,
              int B, int C) {
    const int wave = threadIdx.x >> 5;
    const int lane = threadIdx.x & 31;
    const int row = blockIdx.x * 8 + wave;
    if (row >= B) return;

    const bool hi = (32 + lane) < C;              // lane also owns class 32+lane
    const float* xr = x + (size_t)row * FEAT;
    float l0 = 0.0f, l1 = 0.0f;
    for (int k = 0; k < FEAT; ++k) {
        const float xv = xr[k];
        l0 += xv * Wl[k * C + lane];
        if (hi) l1 += xv * Wl[k * C + 32 + lane];
    }
    l0 += bl[lane];
    l1 = hi ? (l1 + bl[32 + lane]) : -INFINITY;

    float m = fmaxf(l0, l1);
    #pragma unroll
    for (int s = 16; s > 0; s >>= 1) m = fmaxf(m, __shfl_xor(m, s, 32));
    float e = __expf(l0 - m) + (hi ? __expf(l1 - m) : 0.0f);
    #pragma unroll
    for (int s = 16; s > 0; s >>= 1) e += __shfl_xor(e, s, 32);
    const float lse = m + __logf(e);

    out[(size_t)row * C + lane] = l0 - lse;
    if (hi) out[(size_t)row * C + 32 + lane] = l1 - lse;
}

// ---------------------------------------------------------------------------
extern "C" void kernel_launch(void* const* d_in, const int* in_sizes, int n_in,
                              void* d_out, int out_size, void* d_ws, size_t ws_size,
                              hipStream_t stream) {
    const float* features = (const float*)d_in[0];
    const int*   edge_src = (const int*)  d_in[1];
    const int*   edge_dst = (const int*)  d_in[2];
    const int*   batch    = (const int*)  d_in[3];
    const float* W1 = (const float*)d_in[4];
    const float* b1 = (const float*)d_in[5];
    const float* W2 = (const float*)d_in[6];
    const float* b2 = (const float*)d_in[7];
    const float* Wl = (const float*)d_in[8];
    const float* bl = (const float*)d_in[9];

    const int N = in_sizes[0] / FEAT;   // 100000
    const int E = in_sizes[1];          // 1600000
    const int B = in_sizes[3];          // 1024
    const int C = in_sizes[9];          // 40

    // Workspace (no init needed: every word is written before it is read).
    char* ws = (char*)d_ws;
    int*   off  = (int*)ws;                                       // (N+1) ints
    size_t o1   = (((size_t)(N + 1) * sizeof(int)) + 255) & ~(size_t)255;
    float* buf1 = (float*)(ws + o1);                              // N x 128 (in-place conv1)
    float* buf2 = buf1 + (size_t)N * FEAT;                        // B x 128 (in-place conv2)

    // 1) CSR offsets from sorted edge_dst (binary search, no atomics).
    k_off<<<(N + 1 + 255) / 256, 256, 0, stream>>>(edge_dst, off, E, N);
    // 2) conv1 mean-aggregation over all nodes.
    k_agg<false><<<(N + 7) / 8, 256, 0, stream>>>(features, edge_src, off, nullptr, buf1, N);
    // 3) x1 = relu(mean1 @ W1 + b1) in-place, fp32 WMMA.
    k_gemm_bias_relu<<<(N + 127) / 128, 256, 0, stream>>>(buf1, W1, b1, N);
    // 4) conv2 aggregation restricted to the 1024 batch nodes (~100x less work).
    k_agg<true><<<(B + 7) / 8, 256, 0, stream>>>(buf1, edge_src, off, batch, buf2, B);
    // 5) x2 = relu(mean2 @ W2 + b2) in-place, fp32 WMMA.
    k_gemm_bias_relu<<<(B + 127) / 128, 256, 0, stream>>>(buf2, W2, b2, B);
    // 6) logits + log-softmax into d_out.
    k_logits<<<(B + 7) / 8, 256, 0, stream>>>(buf2, Wl, bl, (float*)d_out, B, C);
}